// AdaptiveGaussianFilter_66675072303489
// MI455X (gfx1250) — compile-verified
//
#include <hip/hip_runtime.h>
#include <cstdint>
#include <cstddef>

// Adaptive Gaussian filter, 7x7, per-pixel sigma, reflect padding.
// Exact identities:
//   w_k ∝ exp(-((dr^2+dc^2)/2) * sigma^2) = t^(dr^2) * t^(dc^2),  t = exp(-sigma^2/2)
//   sum_k w_k = (1 + 2t + 2t^4 + 2t^9)^2
// => one v_exp_f32 per pixel + separable 7+7 tap FMA chain; kernel is HBM-bound.
// CDNA5 path: halo tile staged with global_load_async_to_lds_b32 (ASYNCcnt),
// waited with s_wait_asynccnt, then per-lane sliding 7x7 register window.

#define IMG        256
#define MID        3
#define TILE_ROWS  16
#define HALO_ROWS  (TILE_ROWS + 2 * MID)   // 22
#define HALO_COLS  (IMG + 2 * MID)         // 262
#define LDS_STRIDE 264                     // pad vs 64 banks
#define NTHREADS   256

__device__ __forceinline__ int reflect_idx(int i) {
    // jnp.pad mode="reflect": -1 -> 1, 256 -> 254 (edge not repeated)
    i = (i < 0) ? -i : i;
    i = (i > IMG - 1) ? (2 * (IMG - 1) - i) : i;
    return i;
}

__global__ __launch_bounds__(NTHREADS)
void AdaptiveGaussianFilter_66675072303489_kernel(const float* __restrict__ x,
                                                  const float* __restrict__ sigma,
                                                  float* __restrict__ out) {
    __shared__ float xs[HALO_ROWS][LDS_STRIDE];

    const int tid  = threadIdx.x;            // 0..255
    const int r0   = blockIdx.x * TILE_ROWS; // first output row of this tile
    const int img  = blockIdx.y;             // 0..47 (B*C flattened)
    const size_t ibase = (size_t)img * IMG * IMG;

    // ---- Phase 1: async DMA of the halo tile (global -> LDS, no VGPR round trip).
    // Reflect indexing folded into the per-lane global source addresses; lanes
    // cover 256 consecutive columns per step -> fully coalesced HBM requests.
    for (int idx = tid; idx < HALO_ROWS * HALO_COLS; idx += NTHREADS) {
        const int i  = idx / HALO_COLS;
        const int j  = idx - i * HALO_COLS;
        const int gr = reflect_idx(r0 - MID + i);
        const int gc = reflect_idx(j - MID);
        const uint64_t gaddr = (uint64_t)(uintptr_t)(x + ibase + (size_t)gr * IMG + gc);
        // Generic pointers to LDS carry the LDS byte offset in addr[31:0] (ISA 10.2).
        const uint32_t laddr = (uint32_t)(uintptr_t)(&xs[i][j]);
        asm volatile("global_load_async_to_lds_b32 %0, %1, off"
                     :: "v"(laddr), "v"(gaddr) : "memory");
    }
    asm volatile("s_wait_asynccnt 0" ::: "memory");   // our DMAs landed in LDS
    __syncthreads();                                  // everyone's DMAs landed

    // ---- Phase 2: each lane owns one column; slide a 7x7 register window down.
    const int c = tid;                                // LDS col window: [c, c+6]
    float w[7][7];                                    // w[m % 7][*] holds LDS row m

    #pragma unroll
    for (int m = 0; m < 6; ++m) {
        #pragma unroll
        for (int j = 0; j < 7; ++j) w[m][j] = xs[m][c + j];
    }

    #pragma unroll
    for (int k = 0; k < TILE_ROWS; ++k) {
        // bring in LDS row k+6 (full unroll => pure register renaming, no moves)
        #pragma unroll
        for (int j = 0; j < 7; ++j) w[(k + 6) % 7][j] = xs[k + 6][c + j];

        const int gr = r0 + k;
        const size_t off = ibase + (size_t)gr * IMG + c;

        const float sg = sigma[off];
        const float ss = sg * sg;
        const float t  = __expf(-0.5f * ss);   // only transcendental per pixel
        const float t2 = t * t;
        const float a1 = t;                    // t^1
        const float a2 = t2 * t2;              // t^4
        const float a3 = a2 * a2 * t;          // t^9

        // vertical 7-tap pass for each of the 7 columns
        float vs[7];
        #pragma unroll
        for (int j = 0; j < 7; ++j) {
            vs[j] =            w[(k + 3) % 7][j]
                  + a1 * (w[(k + 2) % 7][j] + w[(k + 4) % 7][j])
                  + a2 * (w[(k + 1) % 7][j] + w[(k + 5) % 7][j])
                  + a3 * (w[(k + 0) % 7][j] + w[(k + 6) % 7][j]);
        }
        // horizontal 7-tap pass
        const float num = vs[3] + a1 * (vs[2] + vs[4])
                                + a2 * (vs[1] + vs[5])
                                + a3 * (vs[0] + vs[6]);
        const float l1  = 1.0f + 2.0f * (a1 + a2 + a3);
        out[off] = num / (l1 * l1);
    }
}

extern "C" void kernel_launch(void* const* d_in, const int* in_sizes, int n_in,
                              void* d_out, int out_size, void* d_ws, size_t ws_size,
                              hipStream_t stream) {
    const float* x     = (const float*)d_in[0];
    const float* sigma = (const float*)d_in[1];
    float*       out   = (float*)d_out;

    const int nimg = in_sizes[0] / (IMG * IMG);      // 16*3 = 48
    dim3 grid(IMG / TILE_ROWS, nimg);                // (16, 48)
    AdaptiveGaussianFilter_66675072303489_kernel<<<grid, NTHREADS, 0, stream>>>(x, sigma, out);
}